// CTCLayer_87557203296392
// MI455X (gfx1250) — compile-verified
//
#include <hip/hip_runtime.h>

typedef __attribute__((ext_vector_type(2))) float v2f;
typedef __attribute__((ext_vector_type(8))) float v8f;

#define T_ 512
#define B_ 1024
#define V_ 96
#define L_ 48
#define NEG_INF (-1e30f)

__device__ __forceinline__ float lse2(float a, float b) {
    float m = fmaxf(a, b);
    return m + __logf(__expf(a - m) + __expf(b - m));
}
__device__ __forceinline__ float lse3(float a, float b, float c) {
    float m = fmaxf(fmaxf(a, b), c);
    return m + __logf(__expf(a - m) + __expf(b - m) + __expf(c - m));
}

// -------- Pass 1: per-(t,b) log-sum-exp over V=96, one wave per 16 rows.
// Data is loaded directly in the WMMA f32 16x16x4 A-matrix lane layout:
//   lane m (0..15)  : row m, column pair {4j, 4j+1}
//   lane m+16       : row m, column pair {4j+2, 4j+3}
// After lane-local exp-accumulation, one V_WMMA_F32_16X16X4_F32 with a
// ones B-matrix reduces the 4 K-slots (i.e. both lane halves) per row and
// broadcasts the row sums into the C/D tile.
__global__ __launch_bounds__(32) void ctc_pass1_lse(const float* __restrict__ yp,
                                                    float* __restrict__ lse_out) {
    const int lane = threadIdx.x;
    const long r0  = (long)blockIdx.x * 16;      // first of 16 rows (r = t*B + b)
    const int  m   = lane & 15;
    const int  hi  = lane >> 4;                  // 0: cols {4j,4j+1}, 1: cols {4j+2,4j+3}

    const float* rp = yp + (r0 + m) * V_ + hi * 2;

    float x[24][2];
    float mx = -3.0e38f;
#pragma unroll
    for (int j = 0; j < 24; ++j) {
        v2f v = *(const v2f*)(rp + j * 4);       // 8B-aligned (even element offset)
        x[j][0] = v.x; x[j][1] = v.y;
        mx = fmaxf(mx, fmaxf(v.x, v.y));
    }
    mx = fmaxf(mx, __shfl_xor(mx, 16, 32));      // row max in both lane halves

    v2f a = {0.f, 0.f};
#pragma unroll
    for (int j = 0; j < 24; ++j) {
        a.x += __expf(x[j][0] - mx);
        a.y += __expf(x[j][1] - mx);
    }

    v2f bones = {1.f, 1.f};                      // B = ones(4x16)
    v8f c = {};
    // D[m,n] = sum_k A[m,k] = full row sum of exp, broadcast across n
    c = __builtin_amdgcn_wmma_f32_16x16x4_f32(false, a, false, bones,
                                              (short)0, c, false, false);

    // Extract row sum for row = lane (lanes 0..15):
    // rows 0..7 live in lanes 0..15 element r; rows 8..15 in lanes 16..31.
    const int k = lane & 7;
    float e8 = c[0];
    if (k == 1) e8 = c[1];
    if (k == 2) e8 = c[2];
    if (k == 3) e8 = c[3];
    if (k == 4) e8 = c[4];
    if (k == 5) e8 = c[5];
    if (k == 6) e8 = c[6];
    if (k == 7) e8 = c[7];
    float sh = __shfl(e8, (lane & 15) + 16, 32);
    float rs = (lane & 8) ? sh : e8;

    if (lane < 16) lse_out[r0 + lane] = mx + __logf(rs);
}

// -------- Pass 2: sequential CTC forward DP, one wave32 per example.
// State mapping: s = 4*lane + r, r in {0..3}; S = 97 => lanes 0..24 active.
// Even states never take the skip path, so the only cross-lane term per
// step is alpha[4l-1] (previous lane's a3): one shfl_up per timestep.
__global__ __launch_bounds__(32) void ctc_pass2_alpha(const int* __restrict__ yt,
                                                      const float* __restrict__ yp,
                                                      const float* __restrict__ lse,
                                                      float* __restrict__ loss) {
    const int b    = blockIdx.x;
    const int lane = threadIdx.x;

    int i0 = 2 * lane; if (i0 > L_ - 2) i0 = L_ - 2;   // clamp lanes >= 24
    const int lab0 = yt[b * L_ + i0];                  // labels[2l]   (state 4l+1)
    const int lab1 = yt[b * L_ + i0 + 1];              // labels[2l+1] (state 4l+3)
    const int labm1 = __shfl_up(lab1, 1, 32);          // labels[2l-1]
    const bool skip0 = (lane > 0) && (lab0 != labm1);
    const bool skip1 = (lab1 != lab0);

    const long tstride = (long)B_ * V_;
    const float* row0 = yp + (long)b * V_;

    // t = 0: only s=0 (blank) and s=1 (first label) reachable
    float ls = lse[b];
    float a0 = (lane == 0) ? (row0[0]    - ls) : NEG_INF;
    float a1 = (lane == 0) ? (row0[lab0] - ls) : NEG_INF;
    float a2 = NEG_INF, a3 = NEG_INF;

    for (int t = 1; t < T_; ++t) {
        const float* rt = row0 + (long)t * tstride;
        if (t + 1 < T_)                      // stream next row toward the caches
            __builtin_prefetch(rt + tstride + lane * 3, 0, 3);

        const float lst = lse[t * B_ + b];
        const float ebt = rt[0]    - lst;    // blank emission
        const float e0t = rt[lab0] - lst;
        const float e1t = rt[lab1] - lst;

        float p3 = __shfl_up(a3, 1, 32);     // alpha[4l-1] from lane l-1
        if (lane == 0) p3 = NEG_INF;

        const float n0 = lse2(a0, p3) + ebt;                            // s=4l   (blank)
        const float n1 = lse3(a1, a0, skip0 ? p3 : NEG_INF) + e0t;      // s=4l+1
        const float n2 = lse2(a2, a1) + ebt;                            // s=4l+2 (blank)
        const float n3 = lse3(a3, a2, skip1 ? a1 : NEG_INF) + e1t;      // s=4l+3
        a0 = n0; a1 = n1; a2 = n2; a3 = n3;
    }

    const float v96 = __shfl(a0, 24, 32);    // alpha[S-1]
    const float v95 = __shfl(a3, 23, 32);    // alpha[S-2]
    if (lane == 0) loss[b] = -lse2(v95, v96);
}

// -------- Pass 3: deterministic mean over B (no float atomics).
__global__ __launch_bounds__(256) void ctc_reduce_mean(const float* __restrict__ loss,
                                                       float* __restrict__ out) {
    __shared__ float sm[256];
    const int tid = threadIdx.x;
    float s = 0.f;
    for (int i = tid; i < B_; i += 256) s += loss[i];
    sm[tid] = s;
    __syncthreads();
    for (int off = 128; off > 0; off >>= 1) {
        if (tid < off) sm[tid] += sm[tid + off];
        __syncthreads();
    }
    if (tid == 0) out[0] = sm[0] / (float)B_;
}

extern "C" void kernel_launch(void* const* d_in, const int* in_sizes, int n_in,
                              void* d_out, int out_size, void* d_ws, size_t ws_size,
                              hipStream_t stream) {
    (void)in_sizes; (void)n_in; (void)out_size; (void)ws_size;
    const int*   yt = (const int*)d_in[0];     // y_true [B, L] int32
    const float* yp = (const float*)d_in[1];   // y_pred [T, B, V] float32

    float* lse  = (float*)d_ws;                       // T*B floats (2 MiB)
    float* loss = lse + (size_t)T_ * B_;              // B floats

    ctc_pass1_lse  <<<(T_ * B_) / 16, 32, 0, stream>>>(yp, lse);
    ctc_pass2_alpha<<<B_,             32, 0, stream>>>(yt, yp, lse, loss);
    ctc_reduce_mean<<<1,             256, 0, stream>>>(loss, (float*)d_out);
}